// IIRFilterbank_28613072126394
// MI455X (gfx1250) — compile-verified
//
#include <hip/hip_runtime.h>

typedef __attribute__((ext_vector_type(2))) float v2f;
typedef __attribute__((ext_vector_type(4))) float v4f;
typedef __attribute__((ext_vector_type(8))) float v8f;

#define BATCH 8
#define NFILT 32
#define TIME  160000
#define LBLK  400              // samples per block
#define NBLK  400              // TIME / LBLK
#define NGRP  25               // NBLK / 16 (block-column groups per WMMA tile)
#define NROW  25               // LBLK / 16 (row chunks per block)

// workspace layout (float offsets)
#define WS_BN 0                                  // 32*5 normalized b
#define WS_AN (WS_BN + NFILT*5)                  // 32*4 normalized a[1..4]
#define WS_CC (WS_AN + NFILT*4)                  // 32*LBLK*4 zero-input responses
#define WS_D  (WS_CC + NFILT*LBLK*4)             // 8*32*NBLK*4 block end states
#define WS_S  (WS_D + BATCH*NFILT*NBLK*4)        // 8*32*NBLK*4 block start states

// ---------------- Stage A: per-filter coefficient prep + zero-input responses
__global__ void iir_stageA(const float* __restrict__ bco,
                           const float* __restrict__ aco,
                           float* __restrict__ ws) {
  int f = threadIdx.x;
  if (f >= NFILT) return;
  float inv = 1.0f / aco[f * 5 + 0];
  float an[4], bn[5];
  for (int k = 0; k < 4; ++k) { an[k] = aco[f * 5 + 1 + k] * inv; ws[WS_AN + f * 4 + k] = an[k]; }
  for (int k = 0; k < 5; ++k) { bn[k] = bco[f * 5 + k] * inv;     ws[WS_BN + f * 5 + k] = bn[k]; }
  // Ccoef[i][k]: response at offset i to initial condition y[-1-k] = 1, zero input
  for (int k = 0; k < 4; ++k) {
    float y1 = (k == 0) ? 1.f : 0.f, y2 = (k == 1) ? 1.f : 0.f;
    float y3 = (k == 2) ? 1.f : 0.f, y4 = (k == 3) ? 1.f : 0.f;
    for (int i = 0; i < LBLK; ++i) {
      float yv = -(an[0] * y1 + an[1] * y2 + an[2] * y3 + an[3] * y4);
      ws[WS_CC + (f * LBLK + i) * 4 + k] = yv;
      y4 = y3; y3 = y2; y2 = y1; y1 = yv;
    }
  }
}

// ---------------- Stage B: FIR + zero-state particular solution per block (parallel)
__global__ void __launch_bounds__(256)
iir_stageB(const float* __restrict__ x, const float* __restrict__ ws,
           float* __restrict__ p, float* __restrict__ dbuf) {
  int idx = blockIdx.x * blockDim.x + threadIdx.x;
  if (idx >= BATCH * NFILT * NBLK) return;
  int j = idx % NBLK;
  int f = (idx / NBLK) % NFILT;
  int b = idx / (NBLK * NFILT);
  const float* bn = ws + WS_BN + f * 5;
  const float* an = ws + WS_AN + f * 4;
  float b0 = bn[0], b1 = bn[1], b2 = bn[2], b3 = bn[3], b4 = bn[4];
  float a1 = an[0], a2 = an[1], a3 = an[2], a4 = an[3];
  long t0 = (long)j * LBLK;
  const float* xb = x + (long)b * TIME;
  float xm1 = 0.f, xm2 = 0.f, xm3 = 0.f, xm4 = 0.f;
  if (j > 0) { xm1 = xb[t0 - 1]; xm2 = xb[t0 - 2]; xm3 = xb[t0 - 3]; xm4 = xb[t0 - 4]; }
  float p1 = 0.f, p2 = 0.f, p3 = 0.f, p4 = 0.f;
  float* pout = p + (long)(b * NFILT + f) * TIME + t0;
  for (int i = 0; i < LBLK; ++i) {
    float xi = xb[t0 + i];
    float fir = b0 * xi + b1 * xm1 + b2 * xm2 + b3 * xm3 + b4 * xm4;
    float pv = fir - (a1 * p1 + a2 * p2 + a3 * p3 + a4 * p4);
    pout[i] = pv;
    xm4 = xm3; xm3 = xm2; xm2 = xm1; xm1 = xi;
    p4 = p3; p3 = p2; p2 = p1; p1 = pv;
  }
  float* d = dbuf + (long)((b * NFILT + f) * NBLK + j) * 4;
  d[0] = p1; d[1] = p2; d[2] = p3; d[3] = p4;   // end state (y[L-1..L-4])
}

// ---------------- Stage C: serial 4x4 state scan across blocks (short)
__global__ void iir_stageC(const float* __restrict__ cc,
                           const float* __restrict__ dbuf,
                           float* __restrict__ sbuf) {
  int tid = threadIdx.x;
  if (tid >= BATCH * NFILT) return;
  int f = tid % NFILT;
  float P[4][4];   // Phi[r][k] = Ccoef[L-1-r][k]
  for (int r = 0; r < 4; ++r)
    for (int k = 0; k < 4; ++k)
      P[r][k] = cc[(f * LBLK + (LBLK - 1 - r)) * 4 + k];
  float s0 = 0.f, s1 = 0.f, s2 = 0.f, s3 = 0.f;
  const float* d = dbuf + (long)tid * NBLK * 4;
  float* s = sbuf + (long)tid * NBLK * 4;
  for (int j = 0; j < NBLK; ++j) {
    s[j * 4 + 0] = s0; s[j * 4 + 1] = s1; s[j * 4 + 2] = s2; s[j * 4 + 3] = s3;
    float d0 = d[j * 4 + 0], d1 = d[j * 4 + 1], d2 = d[j * 4 + 2], d3 = d[j * 4 + 3];
    float n0 = d0 + P[0][0] * s0 + P[0][1] * s1 + P[0][2] * s2 + P[0][3] * s3;
    float n1 = d1 + P[1][0] * s0 + P[1][1] * s1 + P[1][2] * s2 + P[1][3] * s3;
    float n2 = d2 + P[2][0] * s0 + P[2][1] * s1 + P[2][2] * s2 + P[2][3] * s3;
    float n3 = d3 + P[3][0] * s0 + P[3][1] * s1 + P[3][2] * s2 + P[3][3] * s3;
    s0 = n0; s1 = n1; s2 = n2; s3 = n3;
  }
}

// ---------------- Stage D: y = Ccoef(16x4) x S(4x16) + P(16x16) via WMMA f32 16x16x4
__global__ void __launch_bounds__(256)
iir_stageD(const float* __restrict__ cc, const float* __restrict__ sbuf, float* y) {
  int w = blockIdx.x * (blockDim.x >> 5) + (threadIdx.x >> 5);
  int lane = threadIdx.x & 31;
  int r = w % NROW;
  int g = (w / NROW) % NGRP;
  int f = (w / (NROW * NGRP)) % NFILT;
  int b = w / (NROW * NGRP * NFILT);
  int i0 = r * 16;            // row (in-block offset) tile start
  int j0 = g * 16;            // block-column tile start
  int lo = lane & 15;
  int koff = (lane >> 4) * 2; // lanes 0-15: K 0,1 ; lanes 16-31: K 2,3

  // A: Ccoef rows i0..i0+15, 4 cols (16x4, f32 -> 2 VGPRs)
  v2f A = *(const v2f*)(cc + ((f * LBLK + i0 + lo) * 4 + koff));
  // B: start states of blocks j0..j0+15 (4x16 -> 2 VGPRs)
  long bf = (long)(b * NFILT + f);
  v2f Bm = *(const v2f*)(sbuf + ((bf * NBLK + j0 + lo) * 4 + koff));
  // C: particular solution tile (16x16 -> 8 VGPRs); m is the fast axis of t
  long base = bf * (long)TIME + (long)(j0 + lo) * LBLK + i0 + (long)(lane >> 4) * 8;
  v4f c0 = *(const v4f*)(y + base);
  v4f c1 = *(const v4f*)(y + base + 4);
  v8f C;
  C[0] = c0[0]; C[1] = c0[1]; C[2] = c0[2]; C[3] = c0[3];
  C[4] = c1[0]; C[5] = c1[1]; C[6] = c1[2]; C[7] = c1[3];

  v8f D = __builtin_amdgcn_wmma_f32_16x16x4_f32(false, A, false, Bm,
                                                (short)0, C, false, false);

  v4f d0, d1;
  d0[0] = D[0]; d0[1] = D[1]; d0[2] = D[2]; d0[3] = D[3];
  d1[0] = D[4]; d1[1] = D[5]; d1[2] = D[6]; d1[3] = D[7];
  *(v4f*)(y + base) = d0;
  *(v4f*)(y + base + 4) = d1;
}

extern "C" void kernel_launch(void* const* d_in, const int* in_sizes, int n_in,
                              void* d_out, int out_size, void* d_ws, size_t ws_size,
                              hipStream_t stream) {
  const float* x = (const float*)d_in[0];
  const float* bco = (const float*)d_in[1];
  const float* aco = (const float*)d_in[2];
  float* y = (float*)d_out;
  float* ws = (float*)d_ws;

  iir_stageA<<<1, 32, 0, stream>>>(bco, aco, ws);

  int nB = BATCH * NFILT * NBLK;            // 102400 threads
  iir_stageB<<<(nB + 255) / 256, 256, 0, stream>>>(x, ws, y, ws + WS_D);

  iir_stageC<<<1, 256, 0, stream>>>(ws + WS_CC, ws + WS_D, ws + WS_S);

  int waves = BATCH * NFILT * NGRP * NROW;  // 160000 tiles, 8 waves/block
  iir_stageD<<<waves / 8, 256, 0, stream>>>(ws + WS_CC, ws + WS_S, y);
}